// Convolutional_40166534152416
// MI455X (gfx1250) — compile-verified
//
#include <hip/hip_runtime.h>

// CDNA5 (gfx1250) fp32 WMMA fragments.
typedef __attribute__((ext_vector_type(2))) float v2f;  // A / B frag: 2 VGPRs
typedef __attribute__((ext_vector_type(8))) float v8f;  // C / D frag: 8 VGPRs

#define H_IN   112
#define W_IN   112
#define C_IN   64
#define H_OUT  110
#define W_OUT  110
#define C_OUT  128
#define TILE_W 16
#define OH_T   4           // output rows per block (4 M-tiles per wave)
#define SLAB_H (OH_T + 2)  // input rows staged: 4 outputs + 2 halo
#define SLAB_W 18          // 16 + 2 halo columns for the 3-wide kernel
#define CS     68          // padded chunk stride (floats): bank = (4*row + c) % 64 -> conflict-free

// One block: 4 oh rows x 16 ow x all 128 output channels. 8 waves; wave w owns
// channel tile [16w,16w+16) and carries FOUR 16x16 fp32 accumulators (one per
// oh row). Inner step consumes 8 K-values: all 8 A fragments are loaded before
// any wmma so the 4 ds_load_2addr_b64 overlap under one wait, then 8
// v_wmma_f32_16x16x4_f32 issue back-to-back across 4 independent acc chains.
__global__ __launch_bounds__(256)
void conv3x3_relu_wmma_f32(const float* __restrict__ x,
                           const float* __restrict__ kern,   // HWIO == B[576][128] row-major
                           const float* __restrict__ bias,
                           float* __restrict__ out) {
  __shared__ float xs[SLAB_H * SLAB_W * CS];   // 6 x 18 x 64ch slab, padded: 29,376 B

  const int tid = threadIdx.x;
  const int ow0 = blockIdx.x * TILE_W;   // 0..96 (last ow-tile partial: 110 = 6*16+14)
  const int oh0 = blockIdx.y * OH_T;     // 0..108 (last oh-tile partial: 110 = 28*4-2)
  const int n   = blockIdx.z;            // 0..31

  // ---- cooperative stage: x[n][oh0+r][ow0+j][0..63], r=0..5, j=0..17 -> LDS ----
  // 108 chunks x 64 floats = 1728 float4 -> coalesced b128 loads; OOB rows/cols zero-filled.
  for (int idx = tid; idx < SLAB_H * SLAB_W * 16; idx += 256) {
    const int q  = idx & 15;             // float4 within the 64-channel chunk
    const int j  = (idx >> 4) % SLAB_W;  // slab column
    const int r  = (idx >> 4) / SLAB_W;  // slab row 0..5
    const int iw = ow0 + j;
    const int ih = oh0 + r;
    float4 v = make_float4(0.f, 0.f, 0.f, 0.f);
    if (iw < W_IN && ih < H_IN) {
      const float* gp = x + ((size_t)((n * H_IN + ih) * W_IN + iw)) * C_IN + q * 4;
      v = *(const float4*)gp;
    }
    *(float4*)&xs[(r * SLAB_W + j) * CS + q * 4] = v;
  }
  __syncthreads();

  const int wave = tid >> 5;             // 0..7 : output-channel tile
  const int lane = tid & 31;
  const int col  = lane & 15;            // N column within tile; also A row (M = ow pixel)
  const int sel  = (lane >> 4) << 1;     // K sub-offset per ISA A/B striping: 0 or 2
  const int colg = wave * 16 + col;      // global output channel

  // B fragment source: B[k][colg] = kern[k*128 + colg]; advances 8 rows per inner step.
  const float* bp = kern + colg + sel * C_OUT;

  v8f acc[OH_T] = {};
  for (int kh = 0; kh < 3; ++kh) {
    for (int kw = 0; kw < 3; ++kw) {
      // A row for this lane = ow pixel 'col' -> slab column col+kw; M-tile t uses slab row kh+t.
      const float* a0 = &xs[((kh + 0) * SLAB_W + col + kw) * CS + sel];
      const float* a1 = a0 + SLAB_W * CS;
      const float* a2 = a1 + SLAB_W * CS;
      const float* a3 = a2 + SLAB_W * CS;
#pragma unroll
      for (int c0 = 0; c0 < C_IN; c0 += 8) {
        // --- all loads first: 8 A fragments (4x ds_load_2addr_b64) + 2 B fragments ---
        const v2f a0l = *(const v2f*)(a0 + c0);
        const v2f a0h = *(const v2f*)(a0 + c0 + 4);
        const v2f a1l = *(const v2f*)(a1 + c0);
        const v2f a1h = *(const v2f*)(a1 + c0 + 4);
        const v2f a2l = *(const v2f*)(a2 + c0);
        const v2f a2h = *(const v2f*)(a2 + c0 + 4);
        const v2f a3l = *(const v2f*)(a3 + c0);
        const v2f a3h = *(const v2f*)(a3 + c0 + 4);
        v2f bl, bh;                    // B 4x16 f32: VGPR0 rows k0/k0+2, VGPR1 rows k0+1/k0+3
        bl.x = bp[0];
        bl.y = bp[C_OUT];
        bh.x = bp[4 * C_OUT];
        bh.y = bp[5 * C_OUT];
        bp += 8 * C_OUT;
        // --- 8 wmma, 4 independent accumulation chains, no intervening waits ---
        acc[0] = __builtin_amdgcn_wmma_f32_16x16x4_f32(false, a0l, false, bl, (short)0, acc[0], false, false);
        acc[1] = __builtin_amdgcn_wmma_f32_16x16x4_f32(false, a1l, false, bl, (short)0, acc[1], false, false);
        acc[2] = __builtin_amdgcn_wmma_f32_16x16x4_f32(false, a2l, false, bl, (short)0, acc[2], false, false);
        acc[3] = __builtin_amdgcn_wmma_f32_16x16x4_f32(false, a3l, false, bl, (short)0, acc[3], false, false);
        acc[0] = __builtin_amdgcn_wmma_f32_16x16x4_f32(false, a0h, false, bh, (short)0, acc[0], false, false);
        acc[1] = __builtin_amdgcn_wmma_f32_16x16x4_f32(false, a1h, false, bh, (short)0, acc[1], false, false);
        acc[2] = __builtin_amdgcn_wmma_f32_16x16x4_f32(false, a2h, false, bh, (short)0, acc[2], false, false);
        acc[3] = __builtin_amdgcn_wmma_f32_16x16x4_f32(false, a3h, false, bh, (short)0, acc[3], false, false);
      }
    }
  }

  // ---- epilogue: bias + relu + store (C/D layout: VGPR r = M rows r and r+8) ----
  const float bv  = bias[colg];
  const int mbase = (lane >> 4) << 3;    // lanes 16-31 hold M = 8..15
#pragma unroll
  for (int t = 0; t < OH_T; ++t) {
    const int oh = oh0 + t;
    if (oh < H_OUT) {                    // mask the 2 dead rows of the last oh-tile
#pragma unroll
      for (int r = 0; r < 8; ++r) {
        const int ow = ow0 + mbase + r;
        if (ow < W_OUT) {                // mask the partial last ow-tile
          float v = acc[t][r] + bv;
          v = fmaxf(v, 0.0f);
          out[((size_t)((n * H_OUT + oh) * W_OUT + ow)) * C_OUT + colg] = v;
        }
      }
    }
  }
}

extern "C" void kernel_launch(void* const* d_in, const int* in_sizes, int n_in,
                              void* d_out, int out_size, void* d_ws, size_t ws_size,
                              hipStream_t stream) {
  const float* x    = (const float*)d_in[0];   // (32,112,112,64) fp32
  const float* kern = (const float*)d_in[1];   // (3,3,64,128)   fp32, flat == B[576][128]
  const float* bias = (const float*)d_in[2];   // (128,)         fp32
  float* out        = (float*)d_out;           // (32,110,110,128) fp32

  dim3 grid((W_OUT + TILE_W - 1) / TILE_W,     // 7 ow-tiles
            (H_OUT + OH_T - 1) / OH_T,         // 28 oh-tiles
            32);                               // batch
  conv3x3_relu_wmma_f32<<<grid, dim3(256), 0, stream>>>(x, kern, bias, out);
}